// ADNet_66778151518416
// MI455X (gfx1250) — compile-verified
//
#include <hip/hip_runtime.h>
#include <stdint.h>

#define H  1024
#define W  1024
#define HW (H * W)
#define NB 16
#define TILE 32
#define DXF (1.0f / 126.0f)

// Padded LDS tile pitches (col 0 <-> global x = tx0-4, 16B aligned)
#define PT 40   // sTp/sU/sV pitch
#define P1 34   // sT1 pitch

// ---------------------------------------------------------------------------
// Async global -> LDS copies (CDNA5 gfx1250, VGLOBAL GVS form, ASYNCcnt).
//   global_load_async_to_lds_b32/b128  vdst(LDS byte addr), voffset, saddr
// Low 32 bits of a generic pointer to __shared__ are the LDS byte offset.
// ---------------------------------------------------------------------------
__device__ __forceinline__ void async_ld_f32(const float* lds_dst,
                                             const float* gbase,
                                             unsigned byte_off) {
    unsigned lds_addr = (unsigned)(uintptr_t)lds_dst;
    asm volatile("global_load_async_to_lds_b32 %0, %1, %2"
                 :
                 : "v"(lds_addr), "v"(byte_off), "s"(gbase)
                 : "memory");
}

__device__ __forceinline__ void async_ld_b128(const float* lds_dst,
                                              const float* gbase,
                                              unsigned byte_off) {
    unsigned lds_addr = (unsigned)(uintptr_t)lds_dst;
    asm volatile("global_load_async_to_lds_b128 %0, %1, %2"
                 :
                 : "v"(lds_addr), "v"(byte_off), "s"(gbase)
                 : "memory");
}

__device__ __forceinline__ void wait_async_all() {
    asm volatile("s_wait_asynccnt 0x0" ::: "memory");
}

// ---------------------------------------------------------------------------
// Kernel 1: init scratch (harness poisons ws once, never re-poisons)
// ---------------------------------------------------------------------------
__global__ void k_init(unsigned* ws) {
    if (threadIdx.x == 0) ws[0] = 0u;
}

// ---------------------------------------------------------------------------
// Kernel 2: uv_mag = max |u|,|v|. u,v are channels 0,1 -> contiguous 2*HW
// run at b*4*HW. float4 grid-stride, wave32 shuffle, uint-bit atomicMax.
// ---------------------------------------------------------------------------
__global__ __launch_bounds__(256) void k_absmax(const float4* __restrict__ in4,
                                                unsigned* __restrict__ ws) {
    const unsigned n4 = (unsigned)(NB * 2 * HW) / 4u;  // 8M float4
    float m = 0.0f;
    for (unsigned i = blockIdx.x * blockDim.x + threadIdx.x; i < n4;
         i += gridDim.x * blockDim.x) {
        unsigned e   = i << 2;
        unsigned b   = e >> 21;                 // / (2*HW)
        unsigned r   = e & ((1u << 21) - 1u);
        unsigned src = (b << 22) + r;           // b*4*HW + r
        float4 x = in4[src >> 2];
        m = fmaxf(m, fmaxf(fmaxf(fabsf(x.x), fabsf(x.y)),
                           fmaxf(fabsf(x.z), fabsf(x.w))));
    }
    #pragma unroll
    for (int off = 16; off > 0; off >>= 1)
        m = fmaxf(m, __shfl_xor(m, off, 32));
    if ((threadIdx.x & 31) == 0)
        atomicMax(ws, __float_as_uint(m));
}

// ---------------------------------------------------------------------------
// Kernel 3: fused upwind advection + 9-pt laplacian per 32x32 tile.
// Interior tiles: b128 async DMA (aligned, padded rows). Border tiles:
// per-element clamped b32 async DMA (== replicate padding).
// ---------------------------------------------------------------------------
__global__ __launch_bounds__(256) void k_fused(const float* __restrict__ in,
                                               float* __restrict__ out,
                                               const unsigned* __restrict__ ws) {
    __shared__ __align__(16) float sTp[36 * PT];  // T_prev: rows ty0-2..+33, col0 = tx0-4
    __shared__ __align__(16) float sU [34 * PT];  // u:      rows ty0-1..+32
    __shared__ __align__(16) float sV [34 * PT];  // v
    __shared__ __align__(16) float sT1[34 * P1];  // advected T (halo 1)

    const int tid = threadIdx.x;
    const int b   = blockIdx.z;
    const int tx0 = blockIdx.x * TILE;
    const int ty0 = blockIdx.y * TILE;

    const float* ub = in + (size_t)(b * 4 + 0) * HW;
    const float* vb = in + (size_t)(b * 4 + 1) * HW;
    const float* tb = in + (size_t)(b * 4 + 2) * HW;
    const float* qb = in + (size_t)(b * 4 + 3) * HW;

    const float inv_dx = 126.0f;
    const float uv_mag = __uint_as_float(ws[0]);
    const float dt_adv = 0.5f * 0.1f * DXF / uv_mag;
    const float dt_dif = 0.25f * DXF * DXF;       // 0.5*(dx^2)^2 / (2*dx^2)
    const float dt     = fminf(dt_adv, dt_dif);

    const bool interior = (blockIdx.x >= 1) & (blockIdx.x <= (W / TILE - 2)) &
                          (blockIdx.y >= 1) & (blockIdx.y <= (H / TILE - 2));

    if (interior) {
        // ---- fast path: 16B-aligned b128 async DMA, 40-wide padded rows ----
        const unsigned tbase = (unsigned)(((ty0 - 2) << 10) + (tx0 - 4)) * 4u;
        for (int i = tid; i < 36 * 10; i += 256) {            // T_prev: 36 rows x 10 segs
            int row = i / 10, seg = i - row * 10;
            async_ld_b128(&sTp[row * PT + seg * 4], tb,
                          tbase + (unsigned)(row << 12) + (unsigned)(seg << 4));
        }
        const unsigned ubase = (unsigned)(((ty0 - 1) << 10) + (tx0 - 4)) * 4u;
        for (int i = tid; i < 34 * 10; i += 256) {            // u,v: 34 rows x 10 segs
            int row = i / 10, seg = i - row * 10;
            unsigned off = ubase + (unsigned)(row << 12) + (unsigned)(seg << 4);
            async_ld_b128(&sU[row * PT + seg * 4], ub, off);
            async_ld_b128(&sV[row * PT + seg * 4], vb, off);
        }
    } else {
        // ---- border path: per-element clamped b32 (replicate padding) ------
        for (int i = tid; i < 36 * 36; i += 256) {            // cols 2..37 of sTp
            int ly = i / 36, lx = i - ly * 36;
            int gy = min(max(ty0 + ly - 2, 0), H - 1);
            int gx = min(max(tx0 + lx - 2, 0), W - 1);
            async_ld_f32(&sTp[ly * PT + (lx + 2)], tb,
                         (unsigned)((gy << 10) + gx) * 4u);
        }
        for (int i = tid; i < 34 * 34; i += 256) {            // cols 3..36 of sU/sV
            int ly = i / 34, lx = i - ly * 34;
            int gy = min(max(ty0 + ly - 1, 0), H - 1);
            int gx = min(max(tx0 + lx - 1, 0), W - 1);
            unsigned off = (unsigned)((gy << 10) + gx) * 4u;
            async_ld_f32(&sU[ly * PT + (lx + 3)], ub, off);
            async_ld_f32(&sV[ly * PT + (lx + 3)], vb, off);
        }
    }
    wait_async_all();
    __syncthreads();

    // --- phase 1: upwind advection on extended 34x34 tile --------------------
    // point (ly,lx) <-> global (ty0-1+ly, tx0-1+lx); sTp col = lx+3, row = ly+1
    for (int i = tid; i < 34 * 34; i += 256) {
        int ly = i / 34, lx = i - ly * 34;
        const float c = sTp[(ly + 1) * PT + (lx + 3)];
        const float l = sTp[(ly + 1) * PT + (lx + 2)];
        const float r = sTp[(ly + 1) * PT + (lx + 4)];
        const float t = sTp[ ly      * PT + (lx + 3)];
        const float d = sTp[(ly + 2) * PT + (lx + 3)];
        const float uu = sU[ly * PT + (lx + 3)];
        const float vv = sV[ly * PT + (lx + 3)];
        float dTdx = ((uu > 0.0f) ? (c - l) : 0.0f) + ((uu < 0.0f) ? (r - c) : 0.0f);
        float dTdy = ((vv > 0.0f) ? (c - t) : 0.0f) + ((vv < 0.0f) ? (d - c) : 0.0f);
        sT1[ly * P1 + lx] = c + dt * (-(uu * dTdx + vv * dTdy) * inv_dx);
    }
    __syncthreads();

    // --- phase 2: 9-pt laplacian + source; one float4 of output per thread ---
    {
        const int ly  = tid >> 3;            // 0..31
        const int lx4 = (tid & 7) << 2;      // 0,4,...,28
        const int gy  = ty0 + ly;
        const int gx  = tx0 + lx4;
        const float4 rq = *(const float4*)(qb + ((size_t)gy << 10) + gx);
        float o[4];
        #pragma unroll
        for (int j = 0; j < 4; ++j) {
            int s = (ly + 1) * P1 + (lx4 + j + 1);
            float c  = sT1[s];
            float lf = sT1[s - 1],      rf = sT1[s + 1];
            float tf = sT1[s - P1],     df = sT1[s + P1];
            float tl = sT1[s - P1 - 1], tr = sT1[s - P1 + 1];
            float bl = sT1[s + P1 - 1], br = sT1[s + P1 + 1];
            float lap = 0.25f * (tl + 2.0f * tf + tr +
                                 2.0f * lf - 12.0f * c + 2.0f * rf +
                                 bl + 2.0f * df + br) * (inv_dx * inv_dx);
            float raq = (j == 0) ? rq.x : (j == 1) ? rq.y : (j == 2) ? rq.z : rq.w;
            o[j] = c + dt * (lap + raq);
        }
        float4 o4 = make_float4(o[0], o[1], o[2], o[3]);
        *(float4*)(out + (size_t)b * HW + ((size_t)gy << 10) + gx) = o4;
    }

    // --- second output: dt ----------------------------------------------------
    if (b == 0 && blockIdx.x == 0 && blockIdx.y == 0 && tid == 0)
        out[(size_t)NB * HW] = dt;
}

// ---------------------------------------------------------------------------
extern "C" void kernel_launch(void* const* d_in, const int* in_sizes, int n_in,
                              void* d_out, int out_size, void* d_ws, size_t ws_size,
                              hipStream_t stream) {
    const float* in  = (const float*)d_in[0];
    float*       out = (float*)d_out;
    unsigned*    ws  = (unsigned*)d_ws;

    k_init<<<1, 64, 0, stream>>>(ws);
    k_absmax<<<2048, 256, 0, stream>>>((const float4*)in, ws);

    dim3 grid(W / TILE, H / TILE, NB);
    k_fused<<<grid, 256, 0, stream>>>(in, out, ws);
}